// MatrixReasoner_11141145166220
// MI455X (gfx1250) — compile-verified
//
#include <hip/hip_runtime.h>

#define THREADS 256
#define VEC 4
#define TILE (THREADS * VEC)   // 1024 elements per tile per block

// ---------------------------------------------------------------------------
// Device-scope non-returning float atomic max (CDNA5 GLOBAL_ATOMIC_MAX_NUM_F32)
// addr = s[base] + vgpr byte offset (GVS mode). No VDST -> tracked on STOREcnt,
// fire-and-forget into the L2 atomic units. Regular temporal hint: `out` is a
// hot 4 MB array we want resident in L2.
// ---------------------------------------------------------------------------
__device__ __forceinline__ void atomic_max_f32_dev(float* base, unsigned byte_off, float v) {
    asm volatile("global_atomic_max_num_f32 %0, %1, %2 scope:SCOPE_DEV"
                 :
                 : "v"(byte_off), "v"(v), "s"(base)
                 : "memory");
}

// ---------------------------------------------------------------------------
// Async global -> LDS 16-byte copy per lane (CDNA5, ASYNCcnt-tracked).
// vdst = LDS byte address, vaddr = 32-bit global byte offset, saddr = base.
// Non-temporal: vals/rows/cols are streamed exactly once (240 MB) and must not
// evict the hot emb_vec/out working set (8 MB) from the 192 MB L2.
// ---------------------------------------------------------------------------
__device__ __forceinline__ void async_ld16_nt(const void* base, unsigned gbyte_off, unsigned lds_byte_addr) {
    asm volatile("global_load_async_to_lds_b128 %0, %1, %2 th:TH_LOAD_NT"
                 :
                 : "v"(lds_byte_addr), "v"(gbyte_off), "s"(base)
                 : "memory");
}

__global__ void __launch_bounds__(THREADS) zero_kernel(float* __restrict__ out, int n) {
    int i = blockIdx.x * THREADS + threadIdx.x;
    if (i < n) out[i] = 0.0f;
}

__global__ void __launch_bounds__(THREADS) scatter_max_main(
    const float* __restrict__ emb,
    const float* __restrict__ vals,
    const int*   __restrict__ rows,
    const int*   __restrict__ cols,
    float*       __restrict__ out,
    int ntiles)
{
    __shared__ __align__(16) float s_vals[2][TILE];
    __shared__ __align__(16) int   s_rows[2][TILE];
    __shared__ __align__(16) int   s_cols[2][TILE];

    const int tid = threadIdx.x;

    // LDS byte addresses of this thread's private 16B slot in each buffer.
    // Generic pointers to __shared__ carry the LDS offset in their low 32 bits.
    unsigned lv[2], lr[2], lc[2];
    lv[0] = (unsigned)(size_t)&s_vals[0][tid * VEC];
    lv[1] = (unsigned)(size_t)&s_vals[1][tid * VEC];
    lr[0] = (unsigned)(size_t)&s_rows[0][tid * VEC];
    lr[1] = (unsigned)(size_t)&s_rows[1][tid * VEC];
    lc[0] = (unsigned)(size_t)&s_cols[0][tid * VEC];
    lc[1] = (unsigned)(size_t)&s_cols[1][tid * VEC];

    int tile = blockIdx.x;
    int b = 0;

    // Prime buffer 0.
    if (tile < ntiles) {
        unsigned go = ((unsigned)tile * TILE + (unsigned)tid * VEC) * 4u;
        async_ld16_nt(vals, go, lv[0]);
        async_ld16_nt(rows, go, lr[0]);
        async_ld16_nt(cols, go, lc[0]);
    }

    while (tile < ntiles) {
        int next = tile + (int)gridDim.x;
        if (next < ntiles) {
            // Prefetch the next tile into the other buffer, then wait until
            // only those 3 new async ops are outstanding (async loads complete
            // in order -> the current tile's data is in LDS).
            int nb = b ^ 1;
            unsigned go = ((unsigned)next * TILE + (unsigned)tid * VEC) * 4u;
            async_ld16_nt(vals, go, lv[nb]);
            async_ld16_nt(rows, go, lr[nb]);
            async_ld16_nt(cols, go, lc[nb]);
            asm volatile("s_wait_asynccnt 0x3" ::: "memory");
        } else {
            asm volatile("s_wait_asynccnt 0x0" ::: "memory");
        }

        // Each lane reads back exactly the 16B it async-loaded: no barrier
        // needed (ASYNCcnt is per-wave, slots are thread-private).
        float4 v = *(const float4*)&s_vals[b][tid * VEC];
        int4   r = *(const int4*)  &s_rows[b][tid * VEC];
        int4   c = *(const int4*)  &s_cols[b][tid * VEC];

        atomic_max_f32_dev(out, (unsigned)c.x * 4u, emb[r.x] * v.x);
        atomic_max_f32_dev(out, (unsigned)c.y * 4u, emb[r.y] * v.y);
        atomic_max_f32_dev(out, (unsigned)c.z * 4u, emb[r.z] * v.z);
        atomic_max_f32_dev(out, (unsigned)c.w * 4u, emb[r.w] * v.w);

        b ^= 1;
        tile = next;
    }
}

__global__ void __launch_bounds__(THREADS) scatter_max_tail(
    const float* __restrict__ emb,
    const float* __restrict__ vals,
    const int*   __restrict__ rows,
    const int*   __restrict__ cols,
    float*       __restrict__ out,
    int start, int nnz)
{
    int i = start + blockIdx.x * THREADS + threadIdx.x;
    if (i < nnz) {
        float va = __builtin_nontemporal_load(&vals[i]);
        int   ro = __builtin_nontemporal_load(&rows[i]);
        int   co = __builtin_nontemporal_load(&cols[i]);
        atomic_max_f32_dev(out, (unsigned)co * 4u, emb[ro] * va);
    }
}

extern "C" void kernel_launch(void* const* d_in, const int* in_sizes, int n_in,
                              void* d_out, int out_size, void* d_ws, size_t ws_size,
                              hipStream_t stream) {
    const float* emb  = (const float*)d_in[0];
    const float* vals = (const float*)d_in[1];
    const int*   rows = (const int*)d_in[2];
    const int*   cols = (const int*)d_in[3];
    // d_in[4] = rel_id (unused by the math)
    float* out = (float*)d_out;

    const int n_ent = in_sizes[0];
    const int nnz   = in_sizes[1];

    // Identity element for segment-max clamped at 0 (all contribs >= 0).
    zero_kernel<<<(n_ent + THREADS - 1) / THREADS, THREADS, 0, stream>>>(out, n_ent);

    const int ntiles = nnz / TILE;
    if (ntiles > 0) {
        int grid = ntiles < 2048 ? ntiles : 2048;
        scatter_max_main<<<grid, THREADS, 0, stream>>>(emb, vals, rows, cols, out, ntiles);
    }

    const int tail_start = ntiles * TILE;
    const int tail_n = nnz - tail_start;
    if (tail_n > 0) {
        scatter_max_tail<<<(tail_n + THREADS - 1) / THREADS, THREADS, 0, stream>>>(
            emb, vals, rows, cols, out, tail_start, nnz);
    }
}